// STGAN_6373731467926
// MI455X (gfx1250) — compile-verified
//
#include <hip/hip_runtime.h>
#include <hip/hip_bf16.h>

typedef float v2f __attribute__((ext_vector_type(2)));
typedef float v4f __attribute__((ext_vector_type(4)));
typedef float v8f __attribute__((ext_vector_type(8)));

#define BN   4
#define TT   24
#define NN   10000
#define FF   64
#define HIDD 128
#define EE   160000
#define ROWS (BN*NN)      /* 40000 */
#define TILES (ROWS/16)   /* 2500  */
#define XSTR 68           /* padded LDS row stride to dodge bank conflicts */

// ---------------- WMMA helpers (V_WMMA_F32_16X16X4_F32, wave32) ----------------
// C/D 16x16 f32: VGPR r, lanes0-15 -> (M=r, N=lane); lanes16-31 -> (M=r+8, N=lane-16)
// A 16x4 f32:    lane&15 = M row; VGPR v holds K = 4*kk + v + 2*(lane>>4)
// B 4x16  f32:   lane&15 = N col; VGPR v holds K = 4*kk + v + 2*(lane>>4)
__device__ __forceinline__ v8f wmma4(v2f a, v2f b, v8f c) {
  return __builtin_amdgcn_wmma_f32_16x16x4_f32(false, a, false, b, (short)0, c,
                                               false, false);
}

__device__ __forceinline__ v2f fragA_lds(const float* xt, int lane, int kk) {
  int m = lane & 15;
  int K = kk * 4 + ((lane >> 4) << 1);
  v2f a;
  a.x = xt[m * XSTR + K];
  a.y = xt[m * XSTR + K + 1];
  return a;
}

__device__ __forceinline__ v2f fragB(const float* W, int ld, int lane, int kk, int jt) {
  int cc = (lane & 15) + jt * 16;
  int K  = kk * 4 + ((lane >> 4) << 1);
  v2f b;
  b.x = W[(size_t)K * ld + cc];
  b.y = W[(size_t)(K + 1) * ld + cc];
  return b;
}

// ---------------- small utility kernels ----------------
__global__ void k_fill(float* __restrict__ p, float v, long n) {
  long i = (long)blockIdx.x * blockDim.x + threadIdx.x;
  if (i < n) p[i] = v;
}

__global__ void k_degcount(const int* __restrict__ eidx, float* __restrict__ deg) {
  int e = blockIdx.x * blockDim.x + threadIdx.x;
  if (e < EE) atomicAdd(&deg[eidx[EE + e]], 1.0f);
}

__global__ void k_rsqrt(const float* __restrict__ deg, float* __restrict__ dinv, int n) {
  int i = blockIdx.x * blockDim.x + threadIdx.x;
  if (i < n) dinv[i] = rsqrtf(deg[i]);  // deg >= 1 always (self loops)
}

// ---------------- hor_ctx = mean6(horizon) @ W_hor ----------------
__global__ __launch_bounds__(128) void k_horctx(const float* __restrict__ hor,
                                                const float* __restrict__ Whor,
                                                float* __restrict__ HC) {
  __shared__ float lds_a[4][16 * XSTR];
  const int tid = threadIdx.x, wave = tid >> 5, lane = tid & 31;
  const int tile = blockIdx.x * 4 + wave;
  const int b = tile / (NN / 16), n0 = (tile % (NN / 16)) * 16;
  const int cc = lane & 15, hf = lane >> 4;
  float* myx = lds_a[wave];

  for (int i = lane; i < 256; i += 32) {
    int flat = i * 4, row = flat >> 6, k = flat & 63;
    v4f s = {0.f, 0.f, 0.f, 0.f};
    for (int ss = 0; ss < 6; ++ss) {
      const v4f* src = (const v4f*)(hor + (((size_t)b * 6 + ss) * NN + n0) * FF);
      v4f v = src[i];
      s.x += v.x; s.y += v.y; s.z += v.z; s.w += v.w;
    }
    const float inv6 = 1.0f / 6.0f;
    s.x *= inv6; s.y *= inv6; s.z *= inv6; s.w *= inv6;
    *(v4f*)&myx[row * XSTR + k] = s;
  }
  __builtin_amdgcn_wave_barrier();  // in-order DS makes cross-lane RAW safe within a wave

  v2f afr[16];
  for (int kk = 0; kk < 16; ++kk) afr[kk] = fragA_lds(myx, lane, kk);

  const long rowbase = (long)b * NN + n0;
  for (int j = 0; j < 4; ++j) {
    v8f c = {0.f, 0.f, 0.f, 0.f, 0.f, 0.f, 0.f, 0.f};
    for (int kk = 0; kk < 16; ++kk) c = wmma4(afr[kk], fragB(Whor, FF, lane, kk, j), c);
    for (int r = 0; r < 8; ++r)
      HC[(rowbase + r + 8 * hf) * FF + j * 16 + cc] = c[r];
  }
}

// ---------------- fused encode + MLP-accumulate + last-slice export ----------------
// per t: x_t = tanh(hist_t @ W_enc + hor_ctx + b_enc)
//   t < T-1 : acc += x_t @ W1[t*64:(t+1)*64, :]   (acc lives in registers, 16x128/wave)
//   t = T-1 : store x_t to H0 for the GCN
__global__ __launch_bounds__(128) void k_encode(const float* __restrict__ hist,
                                                const float* __restrict__ horctx,
                                                const float* __restrict__ Wenc,
                                                const float* __restrict__ benc,
                                                const float* __restrict__ W1,
                                                float* __restrict__ accb,
                                                float* __restrict__ H0) {
  __shared__ float lds_w1[FF * HIDD];        // 32 KB, per-t W1 slice
  __shared__ float lds_x[4][16 * XSTR];      // per-wave tile (history, then x_t)
  const int tid = threadIdx.x, wave = tid >> 5, lane = tid & 31;
  const int tile = blockIdx.x * 4 + wave;    // exactly 2500 tiles
  const int b = tile / (NN / 16), n0 = (tile % (NN / 16)) * 16;
  const int cc = lane & 15, hf = lane >> 4;
  const long rowg0 = (long)b * NN + n0;
  float* myx = lds_x[wave];

  // hor_ctx + b_enc preloaded in C-fragment layout (t-invariant)
  v8f hc[4];
  for (int j = 0; j < 4; ++j)
    for (int r = 0; r < 8; ++r)
      hc[j][r] = horctx[(rowg0 + r + 8 * hf) * FF + j * 16 + cc] + benc[j * 16 + cc];

  v8f acc[8];
  for (int jj = 0; jj < 8; ++jj) acc[jj] = (v8f){0.f, 0.f, 0.f, 0.f, 0.f, 0.f, 0.f, 0.f};

  for (int t = 0; t < TT; ++t) {
    __syncthreads();  // previous-iteration lds_w1 readers done
    {  // cooperative stage of W1 slice t (64x128 contiguous)
      const v4f* src = (const v4f*)(W1 + (size_t)t * FF * HIDD);
      v4f* dst = (v4f*)lds_w1;
      for (int i = tid; i < FF * HIDD / 4; i += 128) dst[i] = src[i];
    }
    {  // per-wave stage of contiguous 16x64 history tile
      const v4f* src = (const v4f*)(hist + (((size_t)b * TT + t) * NN + n0) * FF);
      for (int i = lane; i < 256; i += 32) {
        int flat = i * 4, row = flat >> 6, k = flat & 63;
        *(v4f*)&myx[row * XSTR + k] = src[i];
      }
    }
    __syncthreads();

    // ---- enc GEMM: ct = tanh(hist_tile @ W_enc + hc) ----
    v2f afr[16];
    for (int kk = 0; kk < 16; ++kk) afr[kk] = fragA_lds(myx, lane, kk);
    v8f ct[4];
    for (int j = 0; j < 4; ++j) {
      v8f c = hc[j];
      for (int kk = 0; kk < 16; ++kk) c = wmma4(afr[kk], fragB(Wenc, FF, lane, kk, j), c);
      for (int r = 0; r < 8; ++r) c[r] = tanhf(c[r]);
      ct[j] = c;
    }

    // ---- transpose D-layout -> A-layout through LDS ----
    __builtin_amdgcn_wave_barrier();
    for (int j = 0; j < 4; ++j)
      for (int r = 0; r < 8; ++r)
        myx[(r + 8 * hf) * XSTR + j * 16 + cc] = ct[j][r];
    __builtin_amdgcn_wave_barrier();

    if (t == TT - 1) {  // uniform branch: export last slice for GCN
      for (int j = 0; j < 4; ++j)
        for (int r = 0; r < 8; ++r)
          H0[(rowg0 + r + 8 * hf) * FF + j * 16 + cc] = ct[j][r];
    } else {            // acc += x_t @ W1_t
      v2f xfr[16];
      for (int kk = 0; kk < 16; ++kk) xfr[kk] = fragA_lds(myx, lane, kk);
      for (int jj = 0; jj < 8; ++jj) {
        v8f c = acc[jj];
        for (int kk = 0; kk < 16; ++kk)
          c = wmma4(xfr[kk], fragB(lds_w1, HIDD, lane, kk, jj), c);
        acc[jj] = c;
      }
    }
  }

  for (int jj = 0; jj < 8; ++jj)
    for (int r = 0; r < 8; ++r)
      accb[(rowg0 + r + 8 * hf) * HIDD + jj * 16 + cc] = acc[jj][r];
}

// ---------------- XW = (H (+bias)) @ W_fc  (40000 x 64 x 64) ----------------
__global__ __launch_bounds__(128) void k_gemm64(const float* __restrict__ A,
                                                const float* __restrict__ W,
                                                const float* __restrict__ bias,
                                                float* __restrict__ Out) {
  __shared__ float lds_a[4][16 * XSTR];
  const int tid = threadIdx.x, wave = tid >> 5, lane = tid & 31;
  const int tile = blockIdx.x * 4 + wave;
  const int cc = lane & 15, hf = lane >> 4;
  const long rowbase = (long)tile * 16;
  float* myx = lds_a[wave];

  const v4f* src = (const v4f*)(A + rowbase * FF);
  for (int i = lane; i < 256; i += 32) {
    int flat = i * 4, row = flat >> 6, k = flat & 63;
    v4f v = src[i];
    if (bias) { v.x += bias[k]; v.y += bias[k + 1]; v.z += bias[k + 2]; v.w += bias[k + 3]; }
    *(v4f*)&myx[row * XSTR + k] = v;
  }
  __builtin_amdgcn_wave_barrier();

  v2f afr[16];
  for (int kk = 0; kk < 16; ++kk) afr[kk] = fragA_lds(myx, lane, kk);

  for (int j = 0; j < 4; ++j) {
    v8f c = {0.f, 0.f, 0.f, 0.f, 0.f, 0.f, 0.f, 0.f};
    for (int kk = 0; kk < 16; ++kk) c = wmma4(afr[kk], fragB(W, FF, lane, kk, j), c);
    for (int r = 0; r < 8; ++r)
      Out[(rowbase + r + 8 * hf) * FF + j * 16 + cc] = c[r];
  }
}

// ---------------- edge scatter: Hdst[:,c,:] += dinv[r]*dinv[c] * XW[:,r,:] ----------------
__global__ void k_scatter(const float* __restrict__ XW, const int* __restrict__ eidx,
                          const float* __restrict__ dinv, float* __restrict__ Hdst) {
  long tid = (long)blockIdx.x * blockDim.x + threadIdx.x;
  const long total = (long)(EE + NN) * BN * 16;
  if (tid >= total) return;
  int f4 = (int)(tid & 15) * 4;
  long t2 = tid >> 4;
  int b = (int)(t2 % BN);
  long e = t2 / BN;
  int r, c;
  if (e < EE) { r = eidx[e]; c = eidx[EE + e]; }
  else        { r = c = (int)(e - EE); }  // self loop
  float nrm = dinv[r] * dinv[c];
  const float* s = XW + ((long)b * NN + r) * FF + f4;
  float* d = Hdst + ((long)b * NN + c) * FF + f4;
  atomicAdd(d + 0, nrm * s[0]);
  atomicAdd(d + 1, nrm * s[1]);
  atomicAdd(d + 2, nrm * s[2]);
  atomicAdd(d + 3, nrm * s[3]);
}

// ---------------- final MLP: out = relu(acc + (h3+b_gcn)@W1_last + b1) @ W2 + b2 ----------------
__global__ __launch_bounds__(128) void k_final(const float* __restrict__ H,
                                               const float* __restrict__ bgcn,
                                               const float* __restrict__ accb,
                                               const float* __restrict__ W1,
                                               const float* __restrict__ b1,
                                               const float* __restrict__ W2,
                                               const float* __restrict__ b2,
                                               float* __restrict__ out) {
  __shared__ float lw[FF * HIDD];
  __shared__ float lb1[HIDD];
  __shared__ float lw2[HIDD];
  const int tid = threadIdx.x;
  const float* W1l = W1 + (size_t)(TT - 1) * FF * HIDD;
  for (int i = tid; i < FF * HIDD; i += 128) lw[i] = W1l[i];
  for (int i = tid; i < HIDD; i += 128) { lb1[i] = b1[i]; lw2[i] = W2[i]; }
  __syncthreads();

  long row = (long)blockIdx.x * 128 + tid;
  if (row >= ROWS) return;
  float h[FF];
  for (int f = 0; f < FF; ++f) h[f] = H[row * FF + f] + bgcn[f];
  const float* ac = accb + row * HIDD;
  float o = b2[0];
  for (int j = 0; j < HIDD; ++j) {
    float s = ac[j] + lb1[j];
    for (int f = 0; f < FF; ++f) s = fmaf(h[f], lw[f * HIDD + j], s);
    o += fmaxf(s, 0.0f) * lw2[j];
  }
  out[row] = o;
}

// ---------------- launcher ----------------
extern "C" void kernel_launch(void* const* d_in, const int* in_sizes, int n_in,
                              void* d_out, int out_size, void* d_ws, size_t ws_size,
                              hipStream_t stream) {
  (void)in_sizes; (void)n_in; (void)out_size; (void)ws_size;
  const float* history = (const float*)d_in[0];
  const float* horizon = (const float*)d_in[1];
  const int*   edges   = (const int*)d_in[2];
  const float* W_enc   = (const float*)d_in[3];
  const float* W_hor   = (const float*)d_in[4];
  const float* b_enc   = (const float*)d_in[5];
  const float* W_fc    = (const float*)d_in[6];
  const float* b_gcn   = (const float*)d_in[7];
  const float* W1      = (const float*)d_in[8];
  const float* b1      = (const float*)d_in[9];
  const float* W2      = (const float*)d_in[10];
  const float* b2      = (const float*)d_in[11];
  float* out = (float*)d_out;

  float* acc  = (float*)d_ws;                       // 40000*128
  float* hctx = acc  + (size_t)ROWS * HIDD;         // 40000*64
  float* H0   = hctx + (size_t)ROWS * FF;
  float* H1   = H0   + (size_t)ROWS * FF;
  float* XW   = H1   + (size_t)ROWS * FF;
  float* deg  = XW   + (size_t)ROWS * FF;           // 10000
  float* dinv = deg  + NN;                          // 10000

  const long hn = (long)ROWS * FF;
  const int hb = (int)((hn + 255) / 256);
  const int scb = (int)(((long)(EE + NN) * BN * 16 + 255) / 256);

  // graph degrees / normalization
  k_fill<<<(NN + 255) / 256, 256, 0, stream>>>(deg, 1.0f, NN);   // self loop
  k_degcount<<<(EE + 255) / 256, 256, 0, stream>>>(edges, deg);
  k_rsqrt<<<(NN + 255) / 256, 256, 0, stream>>>(deg, dinv, NN);

  // horizon context, then fused encode + MLP accumulation
  k_horctx<<<TILES / 4, 128, 0, stream>>>(horizon, W_hor, hctx);
  k_encode<<<TILES / 4, 128, 0, stream>>>(history, hctx, W_enc, b_enc, W1, acc, H0);

  // 3 GCN propagation steps (H0 -> H1 -> H0 -> H1)
  k_gemm64<<<TILES / 4, 128, 0, stream>>>(H0, W_fc, nullptr, XW);
  k_fill<<<hb, 256, 0, stream>>>(H1, 0.0f, hn);
  k_scatter<<<scb, 256, 0, stream>>>(XW, edges, dinv, H1);

  k_gemm64<<<TILES / 4, 128, 0, stream>>>(H1, W_fc, b_gcn, XW);
  k_fill<<<hb, 256, 0, stream>>>(H0, 0.0f, hn);
  k_scatter<<<scb, 256, 0, stream>>>(XW, edges, dinv, H0);

  k_gemm64<<<TILES / 4, 128, 0, stream>>>(H0, W_fc, b_gcn, XW);
  k_fill<<<hb, 256, 0, stream>>>(H1, 0.0f, hn);
  k_scatter<<<scb, 256, 0, stream>>>(XW, edges, dinv, H1);

  // final head
  k_final<<<(ROWS + 127) / 128, 128, 0, stream>>>(H1, b_gcn, acc, W1, b1, W2, b2, out);
}